// MultiGraphGCN_76261439308386
// MI455X (gfx1250) — compile-verified
//
#include <hip/hip_runtime.h>

// ---------------------------------------------------------------------------
// MultiGraphGCN on gfx1250 (MI455X).
// 3-layer GCN, G=4 graphs, N=50000 nodes, E=800000 edges, C=128 channels.
//
// Per layer:   out = D^-1/2 (A + I) D^-1/2 (x @ W) + b
// Restructured: h' = dis * (x@W)  (row scale in GEMM epilogue)
//               agg[d] += h'[s]   (edge scatter, fp32 atomics into L2)
//               out[i] = (agg[i] + h'[i]) * dis[i] + b   (self-loop folded in)
//
// GEMM uses V_WMMA_F32_16X16X4_F32 (full fp32 precision path).
// ---------------------------------------------------------------------------

typedef float v2f __attribute__((ext_vector_type(2)));
typedef float v8f __attribute__((ext_vector_type(8)));

#define GCN_N 50000
#define GCN_E 800000
#define GCN_C 128
#define GCN_G 4

// ---------------- degree / normalization ----------------
__global__ __launch_bounds__(256) void k_deg_init(float* deg, int n) {
    int i = blockIdx.x * 256 + threadIdx.x;
    if (i < n) deg[i] = 1.0f;                       // self-loop
}

__global__ __launch_bounds__(256) void k_deg_count(const long long* __restrict__ dst,
                                                   float* __restrict__ deg, int e) {
    int i = blockIdx.x * 256 + threadIdx.x;
    if (i < e) unsafeAtomicAdd(&deg[(int)dst[i]], 1.0f);
}

__global__ __launch_bounds__(256) void k_deg_to_dis(float* deg, int n) {
    int i = blockIdx.x * 256 + threadIdx.x;
    if (i < n) deg[i] = rsqrtf(deg[i]);             // deg >= 1 always
}

// ---------------- GEMM: H = rowscale(dis) * (X @ W), fp32 WMMA ----------------
// Block: 256 threads = 8 waves. Block handles rows [16*bx, 16*bx+16), all 128 cols.
// Wave w owns cols [16w, 16w+16). K-loop in steps of 4 with v_wmma_f32_16x16x4_f32.
__global__ __launch_bounds__(256)
void k_gemm_rowscale(const float* __restrict__ X,    // [N,128]
                     const float* __restrict__ W,    // [128,128] row-major (k,n)
                     const float* __restrict__ dis,  // [N]
                     float* __restrict__ H,          // [N,128]
                     int n) {
    const int wave = threadIdx.x >> 5;
    const int lane = threadIdx.x & 31;
    const int lm   = lane & 15;     // 0..15
    const int hi   = lane >> 4;     // 0 or 1
    const int m0   = blockIdx.x * 16;
    const int n0   = wave * 16;

    v8f acc = {};
    const float* xrow = X + (size_t)(m0 + lm) * GCN_C;   // A row for this lane
    const float* wcol = W + n0 + lm;                     // B col for this lane

    #pragma unroll 4
    for (int k = 0; k < GCN_C; k += 4) {
        const int ka = k + 2 * hi;
        // A 16x4 fp32 layout: lanes 0-15 hold K=k,k+1; lanes 16-31 hold K=k+2,k+3
        v2f a;
        {
            const float2 av = *(const float2*)(xrow + ka);
            a.x = av.x; a.y = av.y;
        }
        // B 4x16 fp32 layout mirrored: lane = N, VGPR0/1 = the same K pair
        v2f b;
        b.x = wcol[(size_t)ka * GCN_C];
        b.y = wcol[(size_t)(ka + 1) * GCN_C];
        acc = __builtin_amdgcn_wmma_f32_16x16x4_f32(
            /*neg_a=*/false, a, /*neg_b=*/false, b,
            /*c_mod=*/(short)0, acc, /*reuse_a=*/false, /*reuse_b=*/false);
    }

    // C/D layout: VGPR j -> (M = j + 8*hi, N = lm). Apply row scale dis[row].
    const int col = n0 + lm;
    #pragma unroll
    for (int j = 0; j < 8; ++j) {
        const int row = m0 + j + 8 * hi;
        H[(size_t)row * GCN_C + col] = acc[j] * dis[row];
    }
}

// ---------------- zero the aggregation buffer ----------------
__global__ __launch_bounds__(256) void k_zero4(float4* p, int n4) {
    int i = blockIdx.x * 256 + threadIdx.x;
    if (i < n4) p[i] = make_float4(0.f, 0.f, 0.f, 0.f);
}

// ---------------- edge scatter: agg[dst] += h'[src] ----------------
// One wave per edge; each lane handles 4 consecutive channels (float4 of 128).
__global__ __launch_bounds__(256)
void k_scatter(const long long* __restrict__ src,
               const long long* __restrict__ dst,
               const float* __restrict__ H,
               float* __restrict__ AGG, int e) {
    const int edge = blockIdx.x * 8 + (threadIdx.x >> 5);
    if (edge >= e) return;
    const int lane = threadIdx.x & 31;
    const int s = (int)src[edge];
    const int d = (int)dst[edge];
    const float4 v = *(const float4*)(H + (size_t)s * GCN_C + lane * 4);
    float* o = AGG + (size_t)d * GCN_C + lane * 4;
    unsafeAtomicAdd(o + 0, v.x);
    unsafeAtomicAdd(o + 1, v.y);
    unsafeAtomicAdd(o + 2, v.z);
    unsafeAtomicAdd(o + 3, v.w);
}

// ---------------- finalize: out = (agg + h') * dis + b ----------------
__global__ __launch_bounds__(256)
void k_finalize(const float* __restrict__ AGG,
                const float* __restrict__ H,
                const float* __restrict__ dis,
                const float* __restrict__ bias,
                float* __restrict__ OUT, int n) {
    int idx = blockIdx.x * 256 + threadIdx.x;       // over n * (128/4) groups
    if (idx >= n * (GCN_C / 4)) return;
    const int row = idx >> 5;
    const int c4  = (idx & 31) * 4;
    const float ds = dis[row];
    const float4 a = *(const float4*)(AGG + (size_t)row * GCN_C + c4);
    const float4 h = *(const float4*)(H   + (size_t)row * GCN_C + c4);
    const float4 b = *(const float4*)(bias + c4);
    float4 o;
    o.x = (a.x + h.x) * ds + b.x;
    o.y = (a.y + h.y) * ds + b.y;
    o.z = (a.z + h.z) * ds + b.z;
    o.w = (a.w + h.w) * ds + b.w;
    *(float4*)(OUT + (size_t)row * GCN_C + c4) = o;
}

// ---------------------------------------------------------------------------
extern "C" void kernel_launch(void* const* d_in, const int* in_sizes, int n_in,
                              void* d_out, int out_size, void* d_ws, size_t ws_size,
                              hipStream_t stream) {
    const int N = GCN_N, E = GCN_E, C = GCN_C, G = GCN_G;

    const float*     x  = (const float*)d_in[0];
    const long long* ei = (const long long*)d_in[1];
    const float* Wm[3] = { (const float*)d_in[2], (const float*)d_in[4], (const float*)d_in[6] };
    const float* Bv[3] = { (const float*)d_in[3], (const float*)d_in[5], (const float*)d_in[7] };
    float* out = (float*)d_out;

    // workspace: bufA [N*C], bufB [N*C], dis [N]  (~51.4 MB)
    float* bufA = (float*)d_ws;
    float* bufB = bufA + (size_t)N * C;
    float* dis  = bufB + (size_t)N * C;

    const int nThreads = 256;
    const int gridN   = (N + nThreads - 1) / nThreads;
    const int gridE   = (E + nThreads - 1) / nThreads;
    const int gridNC4 = ((N * (C / 4)) + nThreads - 1) / nThreads;
    const int gridGemm = N / 16;            // 50000 / 16 = 3125 exact
    const int gridScat = (E + 7) / 8;       // one wave per edge, 8 waves/block

    for (int g = 0; g < G; ++g) {
        const long long* src = ei + (size_t)g * 2 * E;
        const long long* dst = src + E;

        // dis = rsqrt(1 + in-degree)   (same for all 3 layers)
        k_deg_init  <<<gridN, nThreads, 0, stream>>>(dis, N);
        k_deg_count <<<gridE, nThreads, 0, stream>>>(dst, dis, E);
        k_deg_to_dis<<<gridN, nThreads, 0, stream>>>(dis, N);

        const float* cur = x + (size_t)g * N * C;
        for (int l = 0; l < 3; ++l) {
            const float* Wg = Wm[l] + (size_t)g * C * C;
            const float* bg = Bv[l] + (size_t)g * C;

            // h' = dis * (cur @ W)
            k_gemm_rowscale<<<gridGemm, nThreads, 0, stream>>>(cur, Wg, dis, bufB, N);
            // agg = 0 ; agg[dst] += h'[src]
            k_zero4  <<<gridNC4, nThreads, 0, stream>>>((float4*)bufA, N * (C / 4));
            k_scatter<<<gridScat, nThreads, 0, stream>>>(src, dst, bufB, bufA, E);
            // out = (agg + h') * dis + b    (layer 3 -> d_out, else in-place bufA)
            float* dest = (l == 2) ? (out + (size_t)g * N * C) : bufA;
            k_finalize<<<gridNC4, nThreads, 0, stream>>>(bufA, bufB, dis, bg, dest, N);
            cur = bufA;
        }
    }
}